// PointMSMamba_24369644437816
// MI455X (gfx1250) — compile-verified
//
#include <hip/hip_runtime.h>
#include <math.h>

#define BATCH 8
#define NPTS  2048
#define DIM   384
#define KNN   8
#define EPSB  1e-5f
#define MROWS (BATCH * NPTS)   // 16384

typedef float v2f __attribute__((ext_vector_type(2)));
typedef float v8f __attribute__((ext_vector_type(8)));
typedef int   v4i_b __attribute__((__vector_size__(16)));   // matches builtin's V4i

// ---------------------------------------------------------------------------
// Async global->LDS staging (CDNA5 GLOBAL_LOAD_ASYNC_TO_LDS_B128, ASYNCcnt),
// with a synchronous float4 fallback if the builtin is not declared.
// Builtin signature (from hipcc diagnostic): (v4i AS1*, v4i AS3*, Iimm, Iimm)
// ---------------------------------------------------------------------------
#if defined(__has_builtin)
#  if __has_builtin(__builtin_amdgcn_global_load_async_to_lds_b128)
#    define HAVE_ASYNC_LDS 1
#  endif
#endif
#ifndef HAVE_ASYNC_LDS
#  define HAVE_ASYNC_LDS 0
#endif

__device__ __forceinline__ void cp16(const float* g, float* l) {
#if HAVE_ASYNC_LDS
    __builtin_amdgcn_global_load_async_to_lds_b128(
        (__attribute__((address_space(1))) v4i_b*)g,
        (__attribute__((address_space(3))) v4i_b*)l, 0, 0);
#else
    *reinterpret_cast<float4*>(l) = *reinterpret_cast<const float4*>(g);
#endif
}

__device__ __forceinline__ void wait_async0() {
#if HAVE_ASYNC_LDS
#  if defined(__has_builtin) && __has_builtin(__builtin_amdgcn_s_wait_asynccnt)
    __builtin_amdgcn_s_wait_asynccnt(0);
#  else
    asm volatile("s_wait_asynccnt 0x0" ::: "memory");
#  endif
#endif
}

// ---------------------------------------------------------------------------
// K0: build Wcat (768 x 384):  rows 0..383  = w1[:, 0:384]
//                              rows 384..767 = w1[:, 384:768] - w1[:, 0:384]
// ---------------------------------------------------------------------------
__global__ void build_wcat(const float* __restrict__ w1, float* __restrict__ wcat) {
    int i = blockIdx.x * blockDim.x + threadIdx.x;
    const int total = 2 * DIM * DIM;
    if (i >= total) return;
    int j = i / DIM, c = i % DIM;
    float val;
    if (j < DIM) {
        val = w1[(size_t)j * (2 * DIM) + c];
    } else {
        int d = j - DIM;
        val = w1[(size_t)d * (2 * DIM) + DIM + c] - w1[(size_t)d * (2 * DIM) + c];
    }
    wcat[i] = val;
}

// ---------------------------------------------------------------------------
// K1: KNN top-8 (min squared distance). One query per thread, centers in LDS.
// Grid: (BATCH*NPTS/128) blocks of 128 threads; 16 blocks per batch.
// ---------------------------------------------------------------------------
__global__ __launch_bounds__(128) void knn_kernel(const float* __restrict__ center,
                                                  int* __restrict__ idx) {
    __shared__ float cx[NPTS], cy[NPTS], cz[NPTS];
    const int b = blockIdx.x >> 4;                       // 16 blocks per batch
    const int q = ((blockIdx.x & 15) << 7) + threadIdx.x;
    const float* cb = center + (size_t)b * NPTS * 3;
    for (int i = threadIdx.x; i < NPTS; i += blockDim.x) {
        cx[i] = cb[i * 3 + 0];
        cy[i] = cb[i * 3 + 1];
        cz[i] = cb[i * 3 + 2];
    }
    __syncthreads();

    const float qx = cx[q], qy = cy[q], qz = cz[q];
    float bd[KNN];
    int   bi[KNN];
#pragma unroll
    for (int s = 0; s < KNN; ++s) { bd[s] = 3.4e38f; bi[s] = 0; }
    for (int j = 0; j < NPTS; ++j) {
        float dx = cx[j] - qx, dy = cy[j] - qy, dz = cz[j] - qz;
        float d2 = dx * dx + dy * dy + dz * dz;
        if (d2 < bd[KNN - 1]) {              // branch-guarded sorted insert
            float cd = d2; int ci = j;
#pragma unroll
            for (int s = 0; s < KNN; ++s) {
                if (cd < bd[s]) {
                    float td = bd[s]; int ti = bi[s];
                    bd[s] = cd; bi[s] = ci;
                    cd = td; ci = ti;
                }
            }
        }
    }
    int* op = idx + ((size_t)b * NPTS + q) * KNN;
#pragma unroll
    for (int s = 0; s < KNN; ++s) op[s] = bi[s];
}

// ---------------------------------------------------------------------------
// K2/K4: fp32 WMMA GEMM.  C[m,n] = sum_k A[m,k] * Bm[n,k]   (Bm row-major NxK)
// Block tile 128x64, BK=32, 8 waves (4 along M x 2 along N), wave tile 32x32.
// Double-buffered LDS; async global->LDS prefetch of tile kb+BK overlaps the
// WMMA loop on tile kb. Optional fused BN + LeakyReLU(0.2) epilogue.
// ---------------------------------------------------------------------------
#define BM 128
#define BN 64
#define BK 32
#define LDT 34   // padded LDS row stride (floats): 8B-aligned rows, non-pow2

__global__ __launch_bounds__(256) void gemm_wmma_f32(
    const float* __restrict__ A, const float* __restrict__ Bm,
    float* __restrict__ C, int Kd, int ldc,
    const float* __restrict__ bg, const float* __restrict__ bb,
    const float* __restrict__ bmu, const float* __restrict__ bva)
{
    __shared__ float As[2][BM * LDT];
    __shared__ float Bs[2][BN * LDT];

    const int tid  = threadIdx.x;
    const int wave = tid >> 5;
    const int lane = tid & 31;
    const int hh   = lane >> 4;        // half-wave select
    const int lr   = lane & 15;
    const int wm   = (wave & 3) * 32;  // wave M offset in block tile
    const int wn   = (wave >> 2) * 32; // wave N offset in block tile
    const int bm   = blockIdx.y * BM;
    const int bn   = blockIdx.x * BN;

    v8f acc[2][2];
    acc[0][0] = {}; acc[0][1] = {}; acc[1][0] = {}; acc[1][1] = {};

    auto stage = [&](int buf, int kb) {
        // A block: 128x32 floats = 1024 float4, 4 per thread
#pragma unroll
        for (int r = 0; r < 4; ++r) {
            int lin = tid + r * 256;
            int row = lin >> 3;
            int c4  = (lin & 7) << 2;
            cp16(&A[(size_t)(bm + row) * Kd + kb + c4], &As[buf][row * LDT + c4]);
        }
        // B block: 64x32 floats = 512 float4, 2 per thread
#pragma unroll
        for (int r = 0; r < 2; ++r) {
            int lin = tid + r * 256;
            int row = lin >> 3;
            int c4  = (lin & 7) << 2;
            cp16(&Bm[(size_t)(bn + row) * Kd + kb + c4], &Bs[buf][row * LDT + c4]);
        }
    };

    auto compute = [&](int buf) {
#pragma unroll
        for (int ks = 0; ks < BK; ks += 4) {
            const int kk = ks + 2 * hh;  // ISA 16x4 f32 A layout: half-wave K pair
            v2f a0 = *reinterpret_cast<const v2f*>(&As[buf][(wm +      lr) * LDT + kk]);
            v2f a1 = *reinterpret_cast<const v2f*>(&As[buf][(wm + 16 + lr) * LDT + kk]);
            v2f b0 = *reinterpret_cast<const v2f*>(&Bs[buf][(wn +      lr) * LDT + kk]);
            v2f b1 = *reinterpret_cast<const v2f*>(&Bs[buf][(wn + 16 + lr) * LDT + kk]);
            acc[0][0] = __builtin_amdgcn_wmma_f32_16x16x4_f32(false, a0, false, b0, (short)0, acc[0][0], false, false);
            acc[0][1] = __builtin_amdgcn_wmma_f32_16x16x4_f32(false, a0, false, b1, (short)0, acc[0][1], false, false);
            acc[1][0] = __builtin_amdgcn_wmma_f32_16x16x4_f32(false, a1, false, b0, (short)0, acc[1][0], false, false);
            acc[1][1] = __builtin_amdgcn_wmma_f32_16x16x4_f32(false, a1, false, b1, (short)0, acc[1][1], false, false);
        }
    };

    // Software pipeline: prefetch(kb+BK) || compute(kb)
    stage(0, 0);
    wait_async0();
    __syncthreads();
    int buf = 0;
    for (int kb = BK; kb < Kd; kb += BK) {
        stage(buf ^ 1, kb);      // async copies into the other buffer
        compute(buf);            // WMMA on current buffer
        wait_async0();           // next tile landed
        __syncthreads();         // all waves done reading current buffer
        buf ^= 1;
    }
    compute(buf);

    // Epilogue + store. C layout: VGPR v -> M = v + 8*half, N = lane&15.
#pragma unroll
    for (int j = 0; j < 2; ++j) {
        const int gn = bn + wn + j * 16 + lr;
        float s = 1.0f, sh = 0.0f;
        if (bg) {
            float sv = bg[gn] / sqrtf(bva[gn] + EPSB);
            s  = sv;
            sh = bb[gn] - bmu[gn] * sv;
        }
#pragma unroll
        for (int i = 0; i < 2; ++i) {
#pragma unroll
            for (int v = 0; v < 8; ++v) {
                const int gm = bm + wm + i * 16 + v + 8 * hh;
                float val = acc[i][j][v];
                if (bg) {
                    val = val * s + sh;
                    val = fmaxf(val, 0.2f * val);   // LeakyReLU(0.2)
                }
                C[(size_t)gm * ldc + gn] = val;
            }
        }
    }
}

// ---------------------------------------------------------------------------
// K3: gather neighbors' u rows (+v), fused BN1 + LeakyReLU, max over K.
// One wave per point; lane handles dims {lane*4 + c*128 : c=0..2} as float4.
// ---------------------------------------------------------------------------
__device__ __forceinline__ void ld4(const float* p, float o[4]) {
    float4 t = *reinterpret_cast<const float4*>(p);
    o[0] = t.x; o[1] = t.y; o[2] = t.z; o[3] = t.w;
}
__device__ __forceinline__ void st4(float* p, const float o[4]) {
    float4 t; t.x = o[0]; t.y = o[1]; t.z = o[2]; t.w = o[3];
    *reinterpret_cast<float4*>(p) = t;
}

__global__ __launch_bounds__(256) void gather_max(
    const float* __restrict__ uv, const int* __restrict__ idx,
    const float* __restrict__ bg, const float* __restrict__ bb,
    const float* __restrict__ bmu, const float* __restrict__ bva,
    float* __restrict__ hmax)
{
    const int wave = threadIdx.x >> 5;
    const int lane = threadIdx.x & 31;
    const int m = blockIdx.x * 8 + wave;   // global point row, 0..16383
    const int b = m >> 11;                 // / NPTS
    const int* ip = idx + (size_t)m * KNN;

    float sc[3][4], sf[3][4], vcen[3][4], acc[3][4];
#pragma unroll
    for (int c = 0; c < 3; ++c) {
        const int d = lane * 4 + c * 128;
        float gg[4], beb[4], mm[4], vv[4];
        ld4(&bg[d], gg); ld4(&bb[d], beb); ld4(&bmu[d], mm); ld4(&bva[d], vv);
#pragma unroll
        for (int t = 0; t < 4; ++t) {
            float s = gg[t] / sqrtf(vv[t] + EPSB);
            sc[c][t] = s;
            sf[c][t] = beb[t] - mm[t] * s;
            acc[c][t] = -3.4e38f;
        }
        ld4(&uv[(size_t)m * (2 * DIM) + DIM + d], vcen[c]);
    }

    int rows[KNN];
#pragma unroll
    for (int k = 0; k < KNN; ++k) rows[k] = b * NPTS + ip[k];

    for (int k = 0; k < KNN; ++k) {
#pragma unroll
        for (int c = 0; c < 3; ++c) {
            const int d = lane * 4 + c * 128;
            float u4[4];
            ld4(&uv[(size_t)rows[k] * (2 * DIM) + d], u4);
#pragma unroll
            for (int t = 0; t < 4; ++t) {
                float h = u4[t] + vcen[c][t];
                float a = h * sc[c][t] + sf[c][t];
                a = fmaxf(a, 0.2f * a);           // LeakyReLU before max
                acc[c][t] = fmaxf(acc[c][t], a);
            }
        }
    }
#pragma unroll
    for (int c = 0; c < 3; ++c)
        st4(&hmax[(size_t)m * DIM + lane * 4 + c * 128], acc[c]);
}

// ---------------------------------------------------------------------------
extern "C" void kernel_launch(void* const* d_in, const int* in_sizes, int n_in,
                              void* d_out, int out_size, void* d_ws, size_t ws_size,
                              hipStream_t stream) {
    const float* x      = (const float*)d_in[0];
    const float* center = (const float*)d_in[1];
    const float* w1     = (const float*)d_in[2];
    const float* w2     = (const float*)d_in[3];
    const float* bn1g = (const float*)d_in[4];
    const float* bn1b = (const float*)d_in[5];
    const float* bn1m = (const float*)d_in[6];
    const float* bn1v = (const float*)d_in[7];
    const float* bn2g = (const float*)d_in[8];
    const float* bn2b = (const float*)d_in[9];
    const float* bn2m = (const float*)d_in[10];
    const float* bn2v = (const float*)d_in[11];
    float* out = (float*)d_out;

    char* ws = (char*)d_ws;
    const size_t IDX_B  = (size_t)MROWS * KNN * sizeof(int);          // 512 KB
    const size_t WCAT_B = (size_t)2 * DIM * DIM * sizeof(float);      // 1.125 MB
    const size_t UV_B   = (size_t)MROWS * 2 * DIM * sizeof(float);    // 48 MB
    int*   idx  = (int*)ws;
    float* wcat = (float*)(ws + IDX_B);
    float* uv   = (float*)(ws + IDX_B + WCAT_B);
    float* hmax = (float*)(ws + IDX_B + WCAT_B + UV_B);

    // K0: fold [w1a ; w1b - w1a] into one 768x384 weight
    build_wcat<<<(2 * DIM * DIM + 255) / 256, 256, 0, stream>>>(w1, wcat);
    // K1: KNN indices, one query per thread (128 blocks x 128 threads)
    knn_kernel<<<MROWS / 128, 128, 0, stream>>>(center, idx);
    // K2: uv = x @ Wcat^T   (16384 x 768 x 384), no epilogue
    gemm_wmma_f32<<<dim3((2 * DIM) / BN, MROWS / BM), 256, 0, stream>>>(
        x, wcat, uv, DIM, 2 * DIM, nullptr, nullptr, nullptr, nullptr);
    // K3: gather u rows, +v, BN1+LReLU, max over K neighbors
    gather_max<<<MROWS / 8, 256, 0, stream>>>(uv, idx, bn1g, bn1b, bn1m, bn1v, hmax);
    // K4: out = LReLU(BN2(hmax @ w2^T))   (16384 x 384 x 384), fused epilogue
    gemm_wmma_f32<<<dim3(DIM / BN, MROWS / BM), 256, 0, stream>>>(
        hmax, w2, out, DIM, DIM, bn2g, bn2b, bn2m, bn2v);
}